// Network_67534065762888
// MI455X (gfx1250) — compile-verified
//
#include <hip/hip_runtime.h>
#include <math.h>

typedef __attribute__((ext_vector_type(2))) float v2f;
typedef __attribute__((ext_vector_type(8))) float v8f;
typedef int v4i __attribute__((vector_size(16)));

#define AS1 __attribute__((address_space(1)))
#define AS3 __attribute__((address_space(3)))

#if __has_builtin(__builtin_amdgcn_global_load_async_to_lds_b128)
#define HAVE_ASYNC_LDS 1
#else
#define HAVE_ASYNC_LDS 0
#endif

__device__ __forceinline__ void async_copy16(const float* gp, float* lp) {
#if HAVE_ASYNC_LDS
  // 16 bytes per lane, global -> LDS, bypassing VGPRs (ASYNCcnt-tracked)
  __builtin_amdgcn_global_load_async_to_lds_b128((AS1 v4i*)gp, (AS3 v4i*)lp, 0, 0);
#else
  lp[0] = gp[0]; lp[1] = gp[1]; lp[2] = gp[2]; lp[3] = gp[3];
#endif
}

__device__ __forceinline__ void async_wait_all() {
#if HAVE_ASYNC_LDS
#if __has_builtin(__builtin_amdgcn_s_wait_asynccnt)
  __builtin_amdgcn_s_wait_asynccnt(0);
#else
  asm volatile("s_wait_asynccnt 0x0" ::: "memory");
#endif
#endif
}

__device__ __forceinline__ v8f wmma4(v2f a, v2f b, v8f c) {
  // D = A(16x4 f32) * B(4x16 f32) + C(16x16 f32)
  return __builtin_amdgcn_wmma_f32_16x16x4_f32(
      /*neg_a=*/false, a, /*neg_b=*/false, b,
      /*c_mod=*/(short)0, c, /*reuse_a=*/false, /*reuse_b=*/false);
}

__device__ __forceinline__ v8f vzero8() {
  v8f v = {0.f, 0.f, 0.f, 0.f, 0.f, 0.f, 0.f, 0.f};
  return v;
}

// ---------------------------------------------------------------------------
// K1: conv 5x5 (1->32) VALID + bias + relu + maxpool(k=4) over W.
// Output laid out as (B,32,62,15) contiguous == (B*62, 480) stage-1 matrix.
// ---------------------------------------------------------------------------
__global__ void conv_relu_pool(const float* __restrict__ x,
                               const float* __restrict__ cw,
                               const float* __restrict__ cb,
                               float* __restrict__ h1) {
  const int blk = blockIdx.x;   // b*62 + n
  const int b = blk / 62, n = blk % 62;
  __shared__ float patch[5 * 64];
  __shared__ float wgt[800];
  __shared__ float bias[32];
  const int tid = threadIdx.x;
  if (tid < 320) patch[tid] = x[b * 66 * 64 + n * 64 + tid];
  for (int i = tid; i < 800; i += 256) wgt[i] = cw[i];
  if (tid < 32) bias[tid] = cb[tid];
  __syncthreads();
  for (int o = tid; o < 480; o += 256) {
    const int c = o / 15, tp = o % 15;
    const float* wp = &wgt[c * 25];
    float m = -1e30f;
#pragma unroll
    for (int j = 0; j < 4; ++j) {
      const int w0 = tp * 4 + j;
      float s = 0.f;
#pragma unroll
      for (int r = 0; r < 5; ++r)
#pragma unroll
        for (int q = 0; q < 5; ++q)
          s = fmaf(wp[r * 5 + q], patch[r * 64 + w0 + q], s);
      m = fmaxf(m, s);
    }
    h1[((b * 32 + c) * 62 + n) * 15 + tp] = fmaxf(m + bias[c], 0.f);
  }
}

// ---------------------------------------------------------------------------
// K2: fused stage-1 GEMM. C = A(63488x480) @ [bn_w(480x64) | g_w(480x32)]
// XA = tanh(C[:, :64] + bn_b), XG = C[:, 64:96]. WMMA f32 16x16x4.
// Double-buffered async global->LDS staging (ASYNCcnt), 992 blocks x 256 thr.
// ---------------------------------------------------------------------------
#define A_TILE (64 * 36)
#define B_TILE (32 * 96)

__device__ __forceinline__ void s1_issue_tile(const float* __restrict__ A,
                                              const float* __restrict__ bnw,
                                              const float* __restrict__ gw,
                                              int mBase, int k0,
                                              float* As, float* Bs, int tid) {
  // A block: 64 rows x 32 cols = 2048 floats, 4 floats/lane, 2 waves of issues
#pragma unroll
  for (int i = 0; i < 2; ++i) {
    const int idx = (tid + i * 256) * 4;
    const int r = idx >> 5, c = idx & 31;
    async_copy16(&A[(size_t)(mBase + r) * 480 + k0 + c], &As[r * 36 + c]);
  }
  // B block: 32 rows x 96 cols = 3072 floats (bn_w | g_w concat)
#pragma unroll
  for (int i = 0; i < 3; ++i) {
    const int idx = (tid + i * 256) * 4;
    const int r = idx / 96, c = idx % 96;
    const float* gp = (c < 64) ? &bnw[(k0 + r) * 64 + c]
                               : &gw[(k0 + r) * 32 + (c - 64)];
    async_copy16(gp, &Bs[r * 96 + c]);
  }
}

__global__ void s1_gemm(const float* __restrict__ A,
                        const float* __restrict__ bnw,
                        const float* __restrict__ bnb,
                        const float* __restrict__ gw,
                        float* __restrict__ XA,
                        float* __restrict__ XG) {
  __shared__ float As[2 * A_TILE];
  __shared__ float Bs[2 * B_TILE];
  const int tid = threadIdx.x;
  const int lane = tid & 31, wid = tid >> 5;
  const int l15 = lane & 15, hi = lane >> 4;
  const int mBase = blockIdx.x * 64;
  v8f acc[3];
#pragma unroll
  for (int t = 0; t < 3; ++t) acc[t] = vzero8();

  // prologue: stage tile 0
  s1_issue_tile(A, bnw, gw, mBase, 0, As, Bs, tid);
  async_wait_all();
  __syncthreads();

  for (int it = 0; it < 15; ++it) {           // 480 / 32 K-steps
    float* Ac = As + (it & 1) * A_TILE;
    float* Bc = Bs + (it & 1) * B_TILE;
    if (it + 1 < 15)                          // stream next tile while computing
      s1_issue_tile(A, bnw, gw, mBase, (it + 1) * 32,
                    As + ((it + 1) & 1) * A_TILE,
                    Bs + ((it + 1) & 1) * B_TILE, tid);
#pragma unroll
    for (int t = 0; t < 3; ++t) {
      const int tt = wid * 3 + t;
      const int tm = tt / 6, tn = tt % 6;
#pragma unroll
      for (int k = 0; k < 32; k += 4) {
        v2f a, bf;
        a.x = Ac[(tm * 16 + l15) * 36 + k + 2 * hi];
        a.y = Ac[(tm * 16 + l15) * 36 + k + 2 * hi + 1];
        bf.x = Bc[(k + 2 * hi) * 96 + tn * 16 + l15];
        bf.y = Bc[(k + 2 * hi + 1) * 96 + tn * 16 + l15];
        acc[t] = wmma4(a, bf, acc[t]);
      }
    }
    async_wait_all();   // next buffer fully landed in LDS
    __syncthreads();    // all waves done reading current buffer
  }
#pragma unroll
  for (int t = 0; t < 3; ++t) {
    const int tt = wid * 3 + t;
    const int tm = tt / 6, tn = tt % 6;
    const int c = tn * 16 + l15;
#pragma unroll
    for (int i = 0; i < 8; ++i) {
      const int row = mBase + tm * 16 + i + 8 * hi;
      const float v = acc[t][i];
      if (c < 64) XA[row * 64 + c] = tanhf(v + bnb[c]);
      else        XG[row * 32 + (c - 64)] = v;
    }
  }
}

// ---------------------------------------------------------------------------
// Shared per-graph adjacency: S = softmax(XA XA^T), diag<-1, sym deg norm,
// out = S_norm @ XG + g_b. (TOPK==N so the top-k mask is all-ones: no-op.)
// XA_s: 64x65 (rows 62,63 zero). XG_s: 64x33 (rows 62,63 zero). S_s: 64x66.
// ---------------------------------------------------------------------------
__device__ __forceinline__ void sogc_adjacency(float* XA_s, float* XG_s,
                                               float* S_s, float* deg_s,
                                               const float* __restrict__ gb,
                                               float* __restrict__ outp) {
  const int tid = threadIdx.x;
  const int lane = tid & 31, wid = tid >> 5;
  const int l15 = lane & 15, hi = lane >> 4;

  // S = XA * XA^T : 16 tiles (4x4), 2 per wave, K=64
#pragma unroll
  for (int t = 0; t < 2; ++t) {
    const int tt = wid * 2 + t;
    const int tm = tt >> 2, tn = tt & 3;
    v8f acc = vzero8();
#pragma unroll
    for (int k = 0; k < 64; k += 4) {
      v2f a, b;
      a.x = XA_s[(tm * 16 + l15) * 65 + k + 2 * hi];
      a.y = XA_s[(tm * 16 + l15) * 65 + k + 2 * hi + 1];
      b.x = XA_s[(tn * 16 + l15) * 65 + k + 2 * hi];      // B[k][n] = XA[n][k]
      b.y = XA_s[(tn * 16 + l15) * 65 + k + 2 * hi + 1];
      acc = wmma4(a, b, acc);
    }
#pragma unroll
    for (int i = 0; i < 8; ++i)
      S_s[(tm * 16 + i + 8 * hi) * 66 + tn * 16 + l15] = acc[i];
  }
  __syncthreads();

  // row softmax (62 cols), diag <- 1, deg^-1/2 with clamp(min=1)
  if (tid < 64) {
    if (tid < 62) {
      float* row = &S_s[tid * 66];
      float m = -1e30f;
      for (int j = 0; j < 62; ++j) m = fmaxf(m, row[j]);
      float s = 0.f;
      for (int j = 0; j < 62; ++j) { float e = __expf(row[j] - m); row[j] = e; s += e; }
      const float inv = 1.f / s;
      for (int j = 0; j < 62; ++j) row[j] *= inv;
      row[tid] = 1.0f;
      float d = 0.f;
      for (int j = 0; j < 62; ++j) d += row[j];
      deg_s[tid] = rsqrtf(fmaxf(d, 1.0f));
    } else {
      deg_s[tid] = 1.0f;   // padded rows: S there is zero anyway
    }
  }
  __syncthreads();

  // symmetric normalization
  for (int idx = tid; idx < 64 * 64; idx += 256) {
    const int r = idx >> 6, c = idx & 63;
    S_s[r * 66 + c] *= deg_s[r] * deg_s[c];
  }
  __syncthreads();

  // OUT = S_norm @ XG : 8 tiles (4x2), 1 per wave, K=64
  {
    const int tm = wid >> 1, tn = wid & 1;
    v8f acc = vzero8();
#pragma unroll
    for (int k = 0; k < 64; k += 4) {
      v2f a, b;
      a.x = S_s[(tm * 16 + l15) * 66 + k + 2 * hi];
      a.y = S_s[(tm * 16 + l15) * 66 + k + 2 * hi + 1];
      b.x = XG_s[(k + 2 * hi) * 33 + tn * 16 + l15];
      b.y = XG_s[(k + 2 * hi + 1) * 33 + tn * 16 + l15];
      acc = wmma4(a, b, acc);
    }
    const int c = tn * 16 + l15;
#pragma unroll
    for (int i = 0; i < 8; ++i) {
      const int r = tm * 16 + i + 8 * hi;
      if (r < 62) outp[r * 32 + c] = acc[i] + gb[c];
    }
  }
}

// K3: stage-1 adjacency (XA/XG precomputed in global by s1_gemm)
__global__ void s1_adj(const float* __restrict__ XA,
                       const float* __restrict__ XG,
                       const float* __restrict__ gb,
                       float* __restrict__ out) {
  __shared__ float smem[10560];
  float* XA_s = smem;            // 64x65
  float* XG_s = smem + 4160;     // 64x33
  float* S_s  = smem + 6272;     // 64x66
  float* deg_s = smem + 10496;   // 64
  const int b = blockIdx.x;
  const int tid = threadIdx.x;
  for (int idx = tid; idx < 64 * 64; idx += 256) {
    const int r = idx >> 6, c = idx & 63;
    XA_s[r * 65 + c] = (r < 62) ? XA[(b * 62 + r) * 64 + c] : 0.f;
  }
  for (int idx = tid; idx < 64 * 32; idx += 256) {
    const int r = idx >> 5, c = idx & 31;
    XG_s[r * 33 + c] = (r < 62) ? XG[(b * 62 + r) * 32 + c] : 0.f;
  }
  __syncthreads();
  sogc_adjacency(XA_s, XG_s, S_s, deg_s, gb, out + b * 62 * 32);
}

// ---------------------------------------------------------------------------
// K4/K5: stages 2 & 3 fully fused per graph (K=32 GEMMs + adjacency in LDS).
// Phase-2 S_s overlays phase-1 X_s/weights region.
// ---------------------------------------------------------------------------
__global__ void sogc_small(const float* __restrict__ X,
                           const float* __restrict__ bnw,
                           const float* __restrict__ bnb,
                           const float* __restrict__ gw,
                           const float* __restrict__ gb,
                           float* __restrict__ out) {
  __shared__ float smem[11520];
  float* XA_s = smem;            // 64x65
  float* XG_s = smem + 4160;     // 64x33
  float* X_s  = smem + 6272;     // 64x33   (phase 1)
  float* bw_s = smem + 8384;     // 32x64   (phase 1)
  float* bb_s = smem + 10432;    // 64      (phase 1)
  float* gw_s = smem + 10496;    // 32x32   (phase 1)
  float* S_s  = smem + 6272;     // 64x66   (phase 2 overlay)
  float* deg_s = smem + 10496;   // 64      (phase 2 overlay)

  const int b = blockIdx.x;
  const int tid = threadIdx.x;
  const int lane = tid & 31, wid = tid >> 5;
  const int l15 = lane & 15, hi = lane >> 4;

  for (int idx = tid; idx < 64 * 32; idx += 256) {
    const int r = idx >> 5, c = idx & 31;
    X_s[r * 33 + c] = (r < 62) ? X[(b * 62 + r) * 32 + c] : 0.f;
  }
  for (int idx = tid; idx < 2048; idx += 256) bw_s[idx] = bnw[idx];
  if (tid < 64) bb_s[tid] = bnb[tid];
  for (int idx = tid; idx < 1024; idx += 256) gw_s[idx] = gw[idx];
  __syncthreads();

  // XA = tanh(X @ bnw + bnb): 16 tiles, 2 per wave, K=32
#pragma unroll
  for (int t = 0; t < 2; ++t) {
    const int tt = wid * 2 + t;
    const int tm = tt >> 2, tn = tt & 3;
    v8f acc = vzero8();
#pragma unroll
    for (int k = 0; k < 32; k += 4) {
      v2f a, bf;
      a.x = X_s[(tm * 16 + l15) * 33 + k + 2 * hi];
      a.y = X_s[(tm * 16 + l15) * 33 + k + 2 * hi + 1];
      bf.x = bw_s[(k + 2 * hi) * 64 + tn * 16 + l15];
      bf.y = bw_s[(k + 2 * hi + 1) * 64 + tn * 16 + l15];
      acc = wmma4(a, bf, acc);
    }
    const int c = tn * 16 + l15;
#pragma unroll
    for (int i = 0; i < 8; ++i) {
      const int r = tm * 16 + i + 8 * hi;
      XA_s[r * 65 + c] = (r < 62) ? tanhf(acc[i] + bb_s[c]) : 0.f;  // keep pad zero
    }
  }
  // XG = X @ gw: 8 tiles, 1 per wave, K=32
  {
    const int tm = wid >> 1, tn = wid & 1;
    v8f acc = vzero8();
#pragma unroll
    for (int k = 0; k < 32; k += 4) {
      v2f a, bf;
      a.x = X_s[(tm * 16 + l15) * 33 + k + 2 * hi];
      a.y = X_s[(tm * 16 + l15) * 33 + k + 2 * hi + 1];
      bf.x = gw_s[(k + 2 * hi) * 32 + tn * 16 + l15];
      bf.y = gw_s[(k + 2 * hi + 1) * 32 + tn * 16 + l15];
      acc = wmma4(a, bf, acc);
    }
    const int c = tn * 16 + l15;
#pragma unroll
    for (int i = 0; i < 8; ++i)
      XG_s[(tm * 16 + i + 8 * hi) * 33 + c] = acc[i];  // pad rows are 0 (X pad 0)
  }
  __syncthreads();
  sogc_adjacency(XA_s, XG_s, S_s, deg_s, gb, out + b * 62 * 32);
}

// K6: maxpool(k=3) over feature dim (32 -> 10), flatten 620, FC to 3 classes.
__global__ void head(const float* __restrict__ H4, const float* __restrict__ fw,
                     const float* __restrict__ fb, float* __restrict__ out) {
  __shared__ float pooled[620];
  __shared__ float part[3][32];
  const int b = blockIdx.x;
  const int tid = threadIdx.x;  // 128
  for (int idx = tid; idx < 620; idx += 128) {
    const int r = idx / 10, t = idx % 10;
    const float* p = &H4[(b * 62 + r) * 32 + t * 3];
    pooled[idx] = fmaxf(p[0], fmaxf(p[1], p[2]));
  }
  __syncthreads();
  if (tid < 96) {
    const int k = tid / 32, l = tid % 32;
    float s = 0.f;
    for (int j = l; j < 620; j += 32) s = fmaf(pooled[j], fw[j * 3 + k], s);
    part[k][l] = s;
  }
  __syncthreads();
  if (tid < 3) {
    float s = fb[tid];
    for (int l = 0; l < 32; ++l) s += part[tid][l];
    out[b * 3 + tid] = s;
  }
}

extern "C" void kernel_launch(void* const* d_in, const int* in_sizes, int n_in,
                              void* d_out, int out_size, void* d_ws, size_t ws_size,
                              hipStream_t stream) {
  const float* x     = (const float*)d_in[0];
  const float* cw    = (const float*)d_in[1];
  const float* cb    = (const float*)d_in[2];
  const float* s1bnw = (const float*)d_in[3];
  const float* s1bnb = (const float*)d_in[4];
  const float* s1gw  = (const float*)d_in[5];
  const float* s1gb  = (const float*)d_in[6];
  const float* s2bnw = (const float*)d_in[7];
  const float* s2bnb = (const float*)d_in[8];
  const float* s2gw  = (const float*)d_in[9];
  const float* s2gb  = (const float*)d_in[10];
  const float* s3bnw = (const float*)d_in[11];
  const float* s3bnb = (const float*)d_in[12];
  const float* s3gw  = (const float*)d_in[13];
  const float* s3gb  = (const float*)d_in[14];
  const float* fw    = (const float*)d_in[15];
  const float* fb    = (const float*)d_in[16];

  float* ws = (float*)d_ws;
  const size_t R = (size_t)1024 * 62;  // 63488 graph-node rows
  float* H1 = ws;                      // (R, 480)
  float* XA = H1 + R * 480;            // (R, 64)
  float* XG = XA + R * 64;             // (R, 32)
  float* H2 = XG + R * 32;             // (R, 32)
  float* H3 = H2 + R * 32;             // (R, 32)
  float* H4 = H3 + R * 32;             // (R, 32)
  float* out = (float*)d_out;

  conv_relu_pool<<<63488, 256, 0, stream>>>(x, cw, cb, H1);
  s1_gemm<<<992, 256, 0, stream>>>(H1, s1bnw, s1bnb, s1gw, XA, XG);
  s1_adj<<<1024, 256, 0, stream>>>(XA, XG, s1gb, H2);
  sogc_small<<<1024, 256, 0, stream>>>(H2, s2bnw, s2bnb, s2gw, s2gb, H3);
  sogc_small<<<1024, 256, 0, stream>>>(H3, s3bnw, s3bnb, s3gw, s3gb, H4);
  head<<<1024, 128, 0, stream>>>(H4, fw, fb, out);
}